// Result_Level_Distill_41094247088579
// MI455X (gfx1250) — compile-verified
//
#include <hip/hip_runtime.h>
#include <hip/hip_bf16.h>
#include <math.h>

typedef __attribute__((ext_vector_type(2))) float v2f;
typedef __attribute__((ext_vector_type(4))) float v4f;
typedef __attribute__((ext_vector_type(8))) float v8f;

#define BQ    8
#define CCLS  10
#define CREG  11
#define HQ    360
#define WQ    360
#define NQ    256
#define HWQ   (HQ*WQ)
#define HW4   (HWQ/4)
#define BHW   (BQ*HWQ)
#define NG    (BHW/4)
#define NBLK  1024
#define VXF   (0.075f*4.0f)
#define EPSG  2.2204460492503131e-16f

// ---------------------------------------------------------------- zero mask (b128 stores)
__global__ void rld_zero_mask(float* __restrict__ mask) {
    int i = blockIdx.x * blockDim.x + threadIdx.x;
    if (i < NG) {
        v4f z = {};
        *(v4f*)(mask + 4 * i) = z;
    }
}

// ---------------------------------------------------------------- gaussian splat
// One block per (batch, box). atomicMax on float-bits (all values >= 0).
__global__ void rld_splat_mask(const float* __restrict__ gt, float* __restrict__ mask) {
    const int b = blockIdx.x / NQ;
    const int n = blockIdx.x % NQ;

    __shared__ int ok;
    if (threadIdx.x == 0) ok = 1;
    __syncthreads();
    // prefix validity: boxes 0..n must all have nonzero element-sum
    if (threadIdx.x <= n) {
        const float* bx = gt + ((size_t)b * NQ + threadIdx.x) * 9;
        float s = 0.0f;
        #pragma unroll
        for (int k = 0; k < 9; ++k) s += bx[k];
        if (s == 0.0f) ok = 0;
    }
    __syncthreads();
    if (!ok) return;

    const float* box = gt + ((size_t)b * NQ + n) * 9;
    const float hgt = box[3] / VXF;   // gaussian_radius(height=w_pix, width=h_pix)
    const float wid = box[4] / VXF;

    // det3d gaussian_radius, min_overlap = 0.5
    const float b1 = hgt + wid;
    const float c1 = wid * hgt * (1.0f - 0.5f) / (1.0f + 0.5f);
    const float r1 = (b1 + sqrtf(fmaxf(b1 * b1 - 4.0f * c1, 0.0f))) / 2.0f;
    const float b2 = 2.0f * (hgt + wid);
    const float c2 = (1.0f - 0.5f) * wid * hgt;
    const float r2 = (b2 + sqrtf(fmaxf(b2 * b2 - 16.0f * c2, 0.0f))) / 2.0f;
    const float a3 = 4.0f * 0.5f;
    const float b3 = -2.0f * 0.5f * (hgt + wid);
    const float c3 = (0.5f - 1.0f) * wid * hgt;
    const float r3 = (b3 + sqrtf(fmaxf(b3 * b3 - 4.0f * a3 * c3, 0.0f))) / 2.0f;

    int ri = (int)fminf(fminf(r1, r2), r3);   // trunc toward zero, like astype(int32)
    if (ri < 0) ri = 0;

    const int cx = (int)((box[0] + 54.0f) / VXF);
    const int cy = (int)((box[1] + 54.0f) / VXF);
    const float rf    = (float)ri;
    const float sigma = (2.0f * rf + 1.0f) / 6.0f;
    const float denom = 2.0f * sigma * sigma;

    const int xmin = max(0, cx - ri), xmax = min(WQ - 1, cx + ri);
    const int ymin = max(0, cy - ri), ymax = min(HQ - 1, cy + ri);
    if (xmin > xmax || ymin > ymax) return;
    const int ww = xmax - xmin + 1;
    const int count = ww * (ymax - ymin + 1);

    unsigned int* mrow = (unsigned int*)(mask + (size_t)b * HWQ);
    for (int i = threadIdx.x; i < count; i += blockDim.x) {
        const int yy = ymin + i / ww;
        const int xx = xmin + i % ww;
        const float dx = (float)(xx - cx);
        const float dy = (float)(yy - cy);
        float g = expf(-(dx * dx + dy * dy) / denom);
        if (g < EPSG) g = 0.0f;                      // gaussian2D eps cutoff
        if (g > 0.0f)                                // |dx|,|dy| <= r holds by window
            atomicMax(&mrow[yy * WQ + xx], __float_as_uint(g));
    }
}

// ---------------------------------------------------------------- fused streaming pass
// Each thread owns 4 consecutive pixels -> per-lane global_load_b128 on every
// channel line. sigmoid/clip are monotone, so max_c clip(sigmoid(t_c/2)) ==
// clip(sigmoid(max_c(t_c)/2)): one expf per pixel instead of ten.
__global__ void rld_fused_reduce(const float* __restrict__ stu_hm,
                                 const float* __restrict__ tea_hm,
                                 const float* __restrict__ stu_reg,
                                 const float* __restrict__ tea_reg,
                                 const float* __restrict__ mask,
                                 float* __restrict__ partials) {
    float s_cls = 0.0f, s_reg = 0.0f, s_msk = 0.0f;
    const int stride = gridDim.x * blockDim.x;
    for (int g = blockIdx.x * blockDim.x + threadIdx.x; g < NG; g += stride) {
        const int idx = 4 * g;
        const int b = idx / HWQ;
        const int p = idx - b * HWQ;

        const v4f m = *(const v4f*)(mask + idx);
        const v4f* sh = (const v4f*)(stu_hm + (size_t)b * CCLS * HWQ + p);
        const v4f* th = (const v4f*)(tea_hm + (size_t)b * CCLS * HWQ + p);
        const v4f* sr = (const v4f*)(stu_reg + (size_t)b * CREG * HWQ + p);
        const v4f* tr = (const v4f*)(tea_reg + (size_t)b * CREG * HWQ + p);
        __builtin_prefetch(sr, 0, 0);   // -> global_prefetch_b8
        __builtin_prefetch(tr, 0, 0);

        v4f m1, tmax;
        m1.x = m1.y = m1.z = m1.w = -INFINITY;
        tmax = m1;
        #pragma unroll
        for (int c = 0; c < CCLS; ++c) {
            const v4f sv = sh[c * HW4];
            const v4f tv = th[c * HW4];
            #pragma unroll
            for (int k = 0; k < 4; ++k) {
                m1[k]   = fmaxf(m1[k], sv[k]);
                tmax[k] = fmaxf(tmax[k], tv[k]);
            }
        }
        v4f acc = {};
        #pragma unroll
        for (int c = 0; c < CREG; ++c) {
            const v4f a = sr[c * HW4];
            const v4f t = tr[c * HW4];
            #pragma unroll
            for (int k = 0; k < 4; ++k) acc[k] += fabsf(a[k] - t[k]);
        }
        #pragma unroll
        for (int k = 0; k < 4; ++k) {
            float sg = 1.0f / (1.0f + expf(-0.5f * tmax[k]));
            sg = fminf(fmaxf(sg, 0.001f), 0.999f);
            s_reg += (acc[k] / 11.0f) * m[k];
            s_cls += fabsf(m1[k] - sg) * m[k];
            s_msk += m[k];
        }
    }

    __shared__ float r0[256], r1[256], r2[256];
    const int t = threadIdx.x;
    r0[t] = s_cls; r1[t] = s_reg; r2[t] = s_msk;
    __syncthreads();
    for (int off = 128; off > 0; off >>= 1) {
        if (t < off) { r0[t] += r0[t + off]; r1[t] += r1[t + off]; r2[t] += r2[t + off]; }
        __syncthreads();
    }
    if (t == 0) {
        partials[blockIdx.x]            = r0[0];
        partials[NBLK + blockIdx.x]     = r1[0];
        partials[2 * NBLK + blockIdx.x] = r2[0];
    }
}

// ---------------------------------------------------------------- final WMMA reduction
// Sum 1024 partials per quantity with V_WMMA_F32_16X16X4_F32: data as A (16x4),
// B = ones(4x16) => D[i][j] = rowsum_i; accumulate 16 chunks via C. Exact & deterministic.
__global__ void rld_final_reduce(const float* __restrict__ partials, float* __restrict__ out) {
    const int lane = threadIdx.x;   // single wave32, EXEC all ones
    float tot[3];
    #pragma unroll
    for (int q = 0; q < 3; ++q) {
        const float* p = partials + q * NBLK;
        v8f acc = {};
        v2f ones; ones.x = 1.0f; ones.y = 1.0f;
        #pragma unroll
        for (int ch = 0; ch < NBLK / 64; ++ch) {
            v2f a;
            a.x = p[ch * 64 + 2 * lane];
            a.y = p[ch * 64 + 2 * lane + 1];
            acc = __builtin_amdgcn_wmma_f32_16x16x4_f32(
                false, a, false, ones, (short)0, acc, false, false);
        }
        // lanes 0-15 hold rows M=0..7, lanes 16-31 hold rows M=8..15 (replicated over j)
        float s = acc[0] + acc[1] + acc[2] + acc[3] + acc[4] + acc[5] + acc[6] + acc[7];
        s += __shfl_xor(s, 16, 32);
        tot[q] = s;
    }
    if (lane == 0) {
        const float wgt = tot[2] + 0.0001f;
        out[0] = tot[0] / wgt;   // loss_cls_distill
        out[1] = tot[1] / wgt;   // loss_reg_distill
    }
}

extern "C" void kernel_launch(void* const* d_in, const int* in_sizes, int n_in,
                              void* d_out, int out_size, void* d_ws, size_t ws_size,
                              hipStream_t stream) {
    (void)in_sizes; (void)n_in; (void)out_size; (void)ws_size;
    const float* stu_hm  = (const float*)d_in[0];
    const float* tea_hm  = (const float*)d_in[1];
    const float* stu_reg = (const float*)d_in[2];
    const float* tea_reg = (const float*)d_in[3];
    const float* gt      = (const float*)d_in[4];
    float* out  = (float*)d_out;
    float* mask = (float*)d_ws;            // BHW floats
    float* partials = mask + BHW;          // 3*NBLK floats, 16B-aligned

    rld_zero_mask   <<<(NG + 255) / 256, 256, 0, stream>>>(mask);
    rld_splat_mask  <<<BQ * NQ,          256, 0, stream>>>(gt, mask);
    rld_fused_reduce<<<NBLK,             256, 0, stream>>>(stu_hm, tea_hm, stu_reg,
                                                           tea_reg, mask, partials);
    rld_final_reduce<<<1,                 32, 0, stream>>>(partials, out);
}